// CapsuleLayer_47528108097732
// MI455X (gfx1250) — compile-verified
//
#include <hip/hip_runtime.h>
#include <math.h>

// CapsuleLayer forward on MI455X (gfx1250).
// Shapes: B=64, O=32, I=512, D=64 (D_out), M=64 (D_in). All f32.
//
// Phase 1: x_hat[b,o,i,d] = sum_m W[o,i,d,m] * x[b,i,m]
//   -> 16384 independent 64x64x64 f32 GEMMs, done with V_WMMA_F32_16X16X4_F32
//      (full fp32 precision; reference is f32).
// Phase 2: 3 dynamic-routing iterations (softmax over O, weighted sum over I,
//   squash, logit update), all bandwidth-bound passes over x_hat in workspace.
//
// Workspace: x_hat (268MB) + logits (4MB) + out_tmp (0.5MB) + m/Z (0.25MB).

#define B_  64
#define O_  32
#define I_  512
#define D_  64   // D_out
#define M_  64   // D_in

typedef __attribute__((ext_vector_type(2))) float v2f;
typedef __attribute__((ext_vector_type(8))) float v8f;

// ---------------------------------------------------------------------------
// Phase 1: x_hat GEMM.  One block of 128 threads (4 waves) per (o,i).
// Wave w computes rows b = w*16 .. w*16+15, all 64 d columns.
// A (16x4 f32): lane L (L&15)=row M; (L>>4) selects K pair {0,1} vs {2,3}.
// B (4x16 f32): lane L (L&15)=col N; (L>>4) selects K pair. Element [k][n] =
//   W[o,i,d=n, m=k]  (contiguous in m per lane -> float2 loads).
// C/D (16x16 f32, 8 VGPRs): lane L col N=(L&15); VGPR r -> row r + (L>>4)*8.
// ---------------------------------------------------------------------------
__global__ __launch_bounds__(128) void
xhat_gemm_wmma(const float* __restrict__ x, const float* __restrict__ w,
               float* __restrict__ xhat) {
  const int oi   = blockIdx.x;        // 0 .. O*I-1
  const int o    = oi / I_;
  const int i    = oi % I_;
  const int wave = threadIdx.x >> 5;  // 0..3 : batch tile
  const int lane = threadIdx.x & 31;
  const int lrow = lane & 15;
  const int khalf = lane >> 4;        // 0 or 1

  // A: row of X for this lane: x[b=wave*16+lrow, i, :]
  const float* __restrict__ xp =
      x + (size_t)(wave * 16 + lrow) * (I_ * M_) + (size_t)i * M_;
  // B: weight block W[o,i,:,:], row-major [d][m]
  const float* __restrict__ wp = w + ((size_t)(o * I_ + i) * D_) * M_;

  v8f c0 = {}, c1 = {}, c2 = {}, c3 = {};

#pragma unroll 4
  for (int k0 = 0; k0 < M_; k0 += 4) {
    const int kk = k0 + khalf * 2;
    v2f a  = *(const v2f*)(xp + kk);
    v2f b0 = *(const v2f*)(wp + (size_t)(0 * 16 + lrow) * M_ + kk);
    v2f b1 = *(const v2f*)(wp + (size_t)(1 * 16 + lrow) * M_ + kk);
    v2f b2 = *(const v2f*)(wp + (size_t)(2 * 16 + lrow) * M_ + kk);
    v2f b3 = *(const v2f*)(wp + (size_t)(3 * 16 + lrow) * M_ + kk);
    c0 = __builtin_amdgcn_wmma_f32_16x16x4_f32(false, a, false, b0, (short)0, c0, false, false);
    c1 = __builtin_amdgcn_wmma_f32_16x16x4_f32(false, a, false, b1, (short)0, c1, false, false);
    c2 = __builtin_amdgcn_wmma_f32_16x16x4_f32(false, a, false, b2, (short)0, c2, false, false);
    c3 = __builtin_amdgcn_wmma_f32_16x16x4_f32(false, a, false, b3, (short)0, c3, false, false);
  }

  // Store: xhat[((b*O + o)*I + i)*D + d]
#pragma unroll
  for (int r = 0; r < 8; ++r) {
    const int bb = wave * 16 + khalf * 8 + r;
    const size_t base = (((size_t)bb * O_ + o) * I_ + i) * D_;
    xhat[base + 0 * 16 + lrow] = c0[r];
    xhat[base + 1 * 16 + lrow] = c1[r];
    xhat[base + 2 * 16 + lrow] = c2[r];
    xhat[base + 3 * 16 + lrow] = c3[r];
  }
}

// ---------------------------------------------------------------------------
// Phase 2a: out[b,o,d] = squash( sum_i c[b,o,i] * x_hat[b,o,i,d] )
// uniform=1: c = 1/O (first routing iter, softmax of zeros).
// else:      c = exp(blog - m) / Z.
// One 64-thread block per (b,o); thread d accumulates over i (coalesced).
// ---------------------------------------------------------------------------
__global__ __launch_bounds__(64) void
reduce_squash(const float* __restrict__ xhat, const float* __restrict__ blog,
              const float* __restrict__ mbuf, const float* __restrict__ zbuf,
              float* __restrict__ out, int uniform) {
  const int bo = blockIdx.x;
  const int b = bo / O_;
  const int o = bo % O_;
  const int d = threadIdx.x;

  const float* __restrict__ xp = xhat + (((size_t)b * O_ + o) * I_) * D_ + d;
  float s = 0.0f;
  if (uniform) {
    for (int i = 0; i < I_; ++i) s += xp[(size_t)i * D_];
    s *= (1.0f / (float)O_);
  } else {
    const float* __restrict__ bl = blog + ((size_t)b * O_ + o) * I_;
    const float* __restrict__ mp = mbuf + (size_t)b * I_;
    const float* __restrict__ zp = zbuf + (size_t)b * I_;
    for (int i = 0; i < I_; ++i) {
      const float ci = expf(bl[i] - mp[i]) / zp[i];   // uniform across block
      s += ci * xp[(size_t)i * D_];
    }
  }

  __shared__ float red[64];
  red[d] = s * s;
  __syncthreads();
#pragma unroll
  for (int off = 32; off > 0; off >>= 1) {
    if (d < off) red[d] += red[d + off];
    __syncthreads();
  }
  const float n2 = red[0];
  const float n  = sqrtf(n2);
  const float f  = (n2 / (1.0f + n2)) / (n + 1e-8f);
  out[((size_t)b * O_ + o) * D_ + d] = s * f;
}

// ---------------------------------------------------------------------------
// Phase 2b: blog[b,o,i] (+)= dot_d( x_hat[b,o,i,:], out[b,o,:] )
// Grid = B*O*2 blocks of 256 threads; each thread handles one i.
// out[b,o,:] staged in LDS (uniform per block).
// ---------------------------------------------------------------------------
__global__ __launch_bounds__(256) void
logits_update(const float* __restrict__ xhat, const float* __restrict__ out,
              float* __restrict__ blog, int accumulate) {
  const int gid  = blockIdx.x;
  const int half = gid & 1;
  const int bo   = gid >> 1;
  const int b = bo / O_;
  const int o = bo % O_;

  __shared__ float os[D_];
  if (threadIdx.x < D_)
    os[threadIdx.x] = out[((size_t)b * O_ + o) * D_ + threadIdx.x];
  __syncthreads();

  const int i = half * 256 + threadIdx.x;
  const float* __restrict__ xp = xhat + (((size_t)b * O_ + o) * I_ + i) * D_;
  float acc = 0.0f;
#pragma unroll
  for (int d = 0; d < D_; d += 4) {
    const float4 v = *(const float4*)(xp + d);
    acc += v.x * os[d] + v.y * os[d + 1] + v.z * os[d + 2] + v.w * os[d + 3];
  }
  const size_t idx = ((size_t)b * O_ + o) * I_ + i;
  blog[idx] = (accumulate ? blog[idx] : 0.0f) + acc;
}

// ---------------------------------------------------------------------------
// Phase 2c: per-(b,i) softmax stats over O: m = max_o blog, Z = sum_o exp(.-m)
// ---------------------------------------------------------------------------
__global__ __launch_bounds__(256) void
softmax_stats(const float* __restrict__ blog, float* __restrict__ mbuf,
              float* __restrict__ zbuf) {
  const int t = blockIdx.x * 256 + threadIdx.x;  // 0 .. B*I-1
  const int b = t / I_;
  const int i = t % I_;
  const float* __restrict__ bl = blog + (size_t)b * (O_ * I_) + i;
  float mx = -3.0e38f;
#pragma unroll
  for (int o = 0; o < O_; ++o) mx = fmaxf(mx, bl[(size_t)o * I_]);
  float z = 0.0f;
#pragma unroll
  for (int o = 0; o < O_; ++o) z += expf(bl[(size_t)o * I_] - mx);
  mbuf[t] = mx;
  zbuf[t] = z;
}

// ---------------------------------------------------------------------------
extern "C" void kernel_launch(void* const* d_in, const int* in_sizes, int n_in,
                              void* d_out, int out_size, void* d_ws, size_t ws_size,
                              hipStream_t stream) {
  const float* x = (const float*)d_in[0];  // [B, I, M]
  const float* w = (const float*)d_in[1];  // [O, I, D, M]
  float* out = (float*)d_out;              // [B, O, D]

  float* ws   = (float*)d_ws;
  float* xhat = ws;                                        // B*O*I*D
  float* blog = xhat + (size_t)B_ * O_ * I_ * D_;          // B*O*I
  float* otmp = blog + (size_t)B_ * O_ * I_;               // B*O*D
  float* mbuf = otmp + (size_t)B_ * O_ * D_;               // B*I
  float* zbuf = mbuf + (size_t)B_ * I_;                    // B*I

  // Phase 1: x_hat GEMMs (f32 WMMA).
  xhat_gemm_wmma<<<O_ * I_, 128, 0, stream>>>(x, w, xhat);

  // Routing iter 0: c uniform -> out0; b1 = dot(x_hat, out0)
  reduce_squash<<<B_ * O_, 64, 0, stream>>>(xhat, nullptr, nullptr, nullptr, otmp, 1);
  logits_update<<<B_ * O_ * 2, 256, 0, stream>>>(xhat, otmp, blog, 0);

  // Routing iter 1: c = softmax_O(b1) -> out1; b2 = b1 + dot(x_hat, out1)
  softmax_stats<<<(B_ * I_) / 256, 256, 0, stream>>>(blog, mbuf, zbuf);
  reduce_squash<<<B_ * O_, 64, 0, stream>>>(xhat, blog, mbuf, zbuf, otmp, 0);
  logits_update<<<B_ * O_ * 2, 256, 0, stream>>>(xhat, otmp, blog, 1);

  // Routing iter 2 (final): c = softmax_O(b2) -> out (written to d_out)
  softmax_stats<<<(B_ * I_) / 256, 256, 0, stream>>>(blog, mbuf, zbuf);
  reduce_squash<<<B_ * O_, 64, 0, stream>>>(xhat, blog, mbuf, zbuf, out, 0);
}